// DGM_15839839388164
// MI455X (gfx1250) — compile-verified
//
#include <hip/hip_runtime.h>
#include <hip/hip_bf16.h>

// ---------------------------------------------------------------------------
// Sizes (fixed by the reference): N=4096 points, D=256 input dim, H=256.
// ---------------------------------------------------------------------------
#define NPTS 4096
#define DDIM 256
#define HDIM 256

typedef float v2f __attribute__((ext_vector_type(2)));
typedef float v8f __attribute__((ext_vector_type(8)));

// FP32 WMMA: D(16x16,f32) = A(16x4,f32) x B(4x16,f32) + C
// 8-arg pattern: (neg_a, A, neg_b, B, c_mod, C, reuse_a, reuse_b)
__device__ __forceinline__ v8f wmma_f32_k4(v2f a, v2f b, v8f c) {
    return __builtin_amdgcn_wmma_f32_16x16x4_f32(
        false, a, false, b, (short)0, c, false, false);
}

// ---------------------------------------------------------------------------
// Kernel 1: dual GEMM  out_a = x @ Wa , out_b = x @ Wb   ([N,D]x[D,H])
// One 16x16 C-tile per wave, K-loop in steps of 4 (V_WMMA_F32_16X16X4_F32).
// ---------------------------------------------------------------------------
__global__ void dgm_dual_gemm(const float* __restrict__ x,
                              const float* __restrict__ Wa,
                              const float* __restrict__ Wb,
                              float* __restrict__ out_a,
                              float* __restrict__ out_b) {
    const int wave = blockIdx.x * (blockDim.x >> 5) + (threadIdx.x >> 5);
    const int lane = threadIdx.x & 31;
    const int lo = lane & 15;
    const int hi = lane >> 4;           // 0 or 1

    const int tilesPerMat = (NPTS / 16) * (HDIM / 16);   // 4096
    const float* W = (wave < tilesPerMat) ? Wa : Wb;
    float* out     = (wave < tilesPerMat) ? out_a : out_b;
    const int t    = (wave < tilesPerMat) ? wave : wave - tilesPerMat;

    const int mBase = (t / (HDIM / 16)) * 16;
    const int nBase = (t % (HDIM / 16)) * 16;

    v8f acc = {};
    const float* aRow = x + (mBase + lo) * DDIM + 2 * hi;
    const float* bCol = W + nBase + lo;
    for (int k = 0; k < DDIM; k += 4) {
        v2f a, b;
        a[0] = aRow[k + 0];
        a[1] = aRow[k + 1];
        b[0] = bCol[(k + 2 * hi + 0) * HDIM];
        b[1] = bCol[(k + 2 * hi + 1) * HDIM];
        acc = wmma_f32_k4(a, b, acc);
    }
#pragma unroll
    for (int r = 0; r < 8; ++r)
        out[(mBase + r + 8 * hi) * HDIM + nBase + lo] = acc[r];
}

// ---------------------------------------------------------------------------
// Kernel 2: row squared norms of x_aux. One wave per row.
// ---------------------------------------------------------------------------
__global__ void dgm_sqnorm(const float* __restrict__ xa, float* __restrict__ sq) {
    const int row  = blockIdx.x * (blockDim.x >> 5) + (threadIdx.x >> 5);
    const int lane = threadIdx.x & 31;
    float s = 0.f;
    for (int k = lane; k < HDIM; k += 32) {
        float v = xa[row * HDIM + k];
        s += v * v;
    }
#pragma unroll
    for (int m = 16; m > 0; m >>= 1) s += __shfl_xor(s, m, 32);
    if (lane == 0) sq[row] = s;
}

// ---------------------------------------------------------------------------
// Kernel 3: Gram + distance, LDS-staged & double-buffered.
//   probs[i][j] = -sqrt(max(sq_i + sq_j - 2*G_ij, 0)),  G = x_aux @ x_aux^T
//
// Block = 256 threads (8 waves) computes a 64x64 output tile; K chunked at 64.
// A chunk (rows mBase..+63) and B chunk (rows nBase..+63, since B = xa^T) are
// staged into LDS by cooperative GLOBAL_LOAD_ASYNC_TO_LDS_B128 (coalesced:
// 16 consecutive lanes cover one 256B row segment), double-buffered so chunk
// c+1 streams in while chunk c computes; drained with s_wait_asynccnt before
// the block barrier.  LDS rows padded to 68 floats so the 16 fragment lanes
// (stride = row) land in 16 distinct banks.  ~68KB of the 320KB WGP LDS.
// Each wave computes two 16x16 subtiles -> 8x A/B reuse out of LDS.
// ---------------------------------------------------------------------------
#define KCH 64
#define LDS_STRIDE 68
#define CHUNK_F (64 * LDS_STRIDE)     // floats per staged 64x64 chunk

__global__ void dgm_gram_dist(const float* __restrict__ xa,
                              const float* __restrict__ sq,
                              float* __restrict__ probs) {
    __shared__ float smem[4 * CHUNK_F];   // [buf0: A,B][buf1: A,B]

    const int tid  = threadIdx.x;
    const int lane = tid & 31;
    const int lo   = lane & 15;
    const int hi   = lane >> 4;
    const int wave = tid >> 5;            // 0..7

    const int mBase = (blockIdx.x / (NPTS / 64)) * 64;
    const int nBase = (blockIdx.x % (NPTS / 64)) * 64;

    // wave -> two 16x16 subtiles: rows (r0, r0+32), col cC within 64x64 tile
    const int r0 = (wave >> 2) * 16;
    const int r1 = r0 + 32;
    const int cC = (wave & 3) * 16;

    // Issue async copies of A/B 64x64 chunks for K-chunk c into buffer b.
    auto issue = [&](int c, int b) {
        const int kBase = c * KCH;
        const uint32_t aOff = (uint32_t)(b * 2 * CHUNK_F) * 4u;
        const uint32_t bOff = (uint32_t)((b * 2 + 1) * CHUNK_F) * 4u;
#pragma unroll
        for (int s = 0; s < 4; ++s) {
            const int slot = tid + s * 256;       // 1024 b128 slots per chunk
            const int row  = slot >> 4;           // 0..63
            const int col  = (slot & 15) << 2;    // 0..60 step 4 floats
            const uint32_t la =
                aOff + (uint32_t)(row * LDS_STRIDE + col) * 4u;
            const uint64_t ga = (uint64_t)(uintptr_t)(
                xa + (size_t)(mBase + row) * HDIM + kBase + col);
            asm volatile("global_load_async_to_lds_b128 %0, %1, off"
                         :: "v"(la), "v"(ga) : "memory");
        }
#pragma unroll
        for (int s = 0; s < 4; ++s) {
            const int slot = tid + s * 256;
            const int row  = slot >> 4;
            const int col  = (slot & 15) << 2;
            const uint32_t la =
                bOff + (uint32_t)(row * LDS_STRIDE + col) * 4u;
            const uint64_t ga = (uint64_t)(uintptr_t)(
                xa + (size_t)(nBase + row) * HDIM + kBase + col);
            asm volatile("global_load_async_to_lds_b128 %0, %1, off"
                         :: "v"(la), "v"(ga) : "memory");
        }
    };

    v8f acc0 = {};
    v8f acc1 = {};

    issue(0, 0);
    asm volatile("s_wait_asynccnt 0x0" ::: "memory");
    __syncthreads();

    for (int c = 0; c < DDIM / KCH; ++c) {
        const int b = c & 1;
        if (c + 1 < DDIM / KCH) issue(c + 1, b ^ 1);

        const float* aB = &smem[b * 2 * CHUNK_F];
        const float* bB = &smem[(b * 2 + 1) * CHUNK_F];
        const float* aP0 = aB + (r0 + lo) * LDS_STRIDE + 2 * hi;
        const float* aP1 = aB + (r1 + lo) * LDS_STRIDE + 2 * hi;
        const float* bP  = bB + (cC + lo) * LDS_STRIDE + 2 * hi;
#pragma unroll
        for (int kk = 0; kk < KCH; kk += 4) {
            v2f a0, a1, bb;
            a0[0] = aP0[kk + 0]; a0[1] = aP0[kk + 1];
            a1[0] = aP1[kk + 0]; a1[1] = aP1[kk + 1];
            bb[0] = bP [kk + 0]; bb[1] = bP [kk + 1];
            acc0 = wmma_f32_k4(a0, bb, acc0);
            acc1 = wmma_f32_k4(a1, bb, acc1);
        }
        // drain the prefetch, then make buffers consistent across the block
        asm volatile("s_wait_asynccnt 0x0" ::: "memory");
        __syncthreads();
    }

    const int col = nBase + cC + lo;
    const float sqc = sq[col];
#pragma unroll
    for (int r = 0; r < 8; ++r) {
        const int row = mBase + r0 + r + 8 * hi;
        float d2 = sq[row] + sqc - 2.f * acc0[r];
        probs[(size_t)row * NPTS + col] = -sqrtf(fmaxf(d2, 0.f));
    }
#pragma unroll
    for (int r = 0; r < 8; ++r) {
        const int row = mBase + r1 + r + 8 * hi;
        float d2 = sq[row] + sqc - 2.f * acc1[r];
        probs[(size_t)row * NPTS + col] = -sqrtf(fmaxf(d2, 0.f));
    }
}

// ---------------------------------------------------------------------------
// Block reduction helpers (blockDim.x == 256).
// ---------------------------------------------------------------------------
__device__ __forceinline__ float blkSum(float v, float* red) {
    const int tid = threadIdx.x;
    red[tid] = v; __syncthreads();
    for (int s = 128; s > 0; s >>= 1) {
        if (tid < s) red[tid] += red[tid + s];
        __syncthreads();
    }
    float r = red[0]; __syncthreads();
    return r;
}
__device__ __forceinline__ float blkMax(float v, float* red) {
    const int tid = threadIdx.x;
    red[tid] = v; __syncthreads();
    for (int s = 128; s > 0; s >>= 1) {
        if (tid < s) red[tid] = fmaxf(red[tid], red[tid + s]);
        __syncthreads();
    }
    float r = red[0]; __syncthreads();
    return r;
}

// ---------------------------------------------------------------------------
// Kernel 4: per-row LayerNorm stats + entmax-1.5 threshold via bisection.
// One block (256 threads) per row; row lives in LDS (16KB of 320KB).
// entmax15 threshold tau solves sum(max((pn-rowmax)/2 - tau, 0)^2) = 1,
// monotone in tau -> 32 bisection steps reach fp32 precision.
// Emits: mean[i], inv[i] = gamma/(std+eps), T[i] = rowmax + 2*tau
// so that  vprobs[i][j] > 0  <=>  pn_ij > T_i.
// ---------------------------------------------------------------------------
__global__ void dgm_rowstats(const float* __restrict__ probs,
                             const float* __restrict__ gamma,
                             const float* __restrict__ beta,
                             float* __restrict__ meanArr,
                             float* __restrict__ invArr,
                             float* __restrict__ Tarr) {
    __shared__ float row[NPTS];
    __shared__ float red[256];
    const int i = blockIdx.x;
    const int tid = threadIdx.x;
    const float g0 = gamma[0], b0 = beta[0];

    for (int j = tid; j < NPTS; j += 256) row[j] = probs[(size_t)i * NPTS + j];
    __syncthreads();

    float s = 0.f, s2 = 0.f;
    for (int j = tid; j < NPTS; j += 256) { float v = row[j]; s += v; s2 += v * v; }
    const float sum  = blkSum(s,  red);
    const float sum2 = blkSum(s2, red);
    const float mean = sum / (float)NPTS;
    const float var  = fmaxf((sum2 - (float)NPTS * mean * mean) / (float)(NPTS - 1), 0.f);
    const float inv  = g0 / (sqrtf(var) + 1e-6f);

    // pn = gamma*(probs-mean)/(std+eps) + beta ; diagonal forced to -1e-6
    for (int j = tid; j < NPTS; j += 256) row[j] = (row[j] - mean) * inv + b0;
    __syncthreads();
    if (tid == 0) row[i] = -1e-6f;
    __syncthreads();

    float mx = -3.4e38f;
    for (int j = tid; j < NPTS; j += 256) mx = fmaxf(mx, row[j]);
    const float rowmax = blkMax(mx, red);

    // bisection on tau in [-1, 0] (X = (pn-rowmax)/2 has max 0)
    float lo = -1.f, hiB = 0.f;
    for (int it = 0; it < 32; ++it) {
        const float tau = 0.5f * (lo + hiB);
        float f = 0.f;
        for (int j = tid; j < NPTS; j += 256) {
            float t = (row[j] - rowmax) * 0.5f - tau;
            if (t > 0.f) f += t * t;
        }
        const float fs = blkSum(f, red);
        if (fs >= 1.f) lo = tau; else hiB = tau;
    }
    if (tid == 0) {
        const float tau = 0.5f * (lo + hiB);
        meanArr[i] = mean;
        invArr[i]  = inv;
        Tarr[i]    = rowmax + 2.f * tau;
    }
}

// ---------------------------------------------------------------------------
// Kernel 5: adjacency + logprobs.
// adj[i][j] = (pn_ij > T_i) || (pn_ji > T_j); logprobs[i] = row popcount.
// The column reads of probs hit L2 (whole 64MB matrix fits in 192MB L2).
// ---------------------------------------------------------------------------
__global__ void dgm_adj(const float* __restrict__ probs,
                        const float* __restrict__ meanArr,
                        const float* __restrict__ invArr,
                        const float* __restrict__ Tarr,
                        const float* __restrict__ beta,
                        float* __restrict__ adj,
                        float* __restrict__ logp) {
    __shared__ float red[256];
    const int i = blockIdx.x;
    const int tid = threadIdx.x;
    const float b0 = beta[0];
    const float mean_i = meanArr[i], inv_i = invArr[i], T_i = Tarr[i];

    float cnt = 0.f;
    for (int j = tid; j < NPTS; j += 256) {
        float pn_ij, pn_ji;
        if (i == j) {
            pn_ij = -1e-6f; pn_ji = -1e-6f;
        } else {
            pn_ij = (probs[(size_t)i * NPTS + j] - mean_i) * inv_i + b0;
            pn_ji = (probs[(size_t)j * NPTS + i] - meanArr[j]) * invArr[j] + b0;
        }
        const bool a = (pn_ij > T_i) || (pn_ji > Tarr[j]);
        adj[(size_t)i * NPTS + j] = a ? 1.0f : 0.0f;
        cnt += a ? 1.0f : 0.0f;
    }
    const float total = blkSum(cnt, red);
    if (tid == 0) logp[i] = total;
}

// ---------------------------------------------------------------------------
// Host-side launcher.
// d_in: [x, batch, W_enc, W_emb, gamma, beta]
// d_out: [x_ | x_aux | adj | logprobs] flat f32
// d_ws : [probs (N*N) | mean | inv | T | sq]
// ---------------------------------------------------------------------------
extern "C" void kernel_launch(void* const* d_in, const int* in_sizes, int n_in,
                              void* d_out, int out_size, void* d_ws, size_t ws_size,
                              hipStream_t stream) {
    (void)in_sizes; (void)n_in; (void)out_size; (void)ws_size;
    const float* x     = (const float*)d_in[0];
    const float* W_enc = (const float*)d_in[2];
    const float* W_emb = (const float*)d_in[3];
    const float* gamma = (const float*)d_in[4];
    const float* beta  = (const float*)d_in[5];

    float* out_x   = (float*)d_out;                       // N*H
    float* out_xa  = out_x  + NPTS * HDIM;                // N*H
    float* out_adj = out_xa + NPTS * HDIM;                // N*N
    float* out_lp  = out_adj + (size_t)NPTS * NPTS;       // N

    float* probs = (float*)d_ws;                          // N*N
    float* meanA = probs + (size_t)NPTS * NPTS;
    float* invA  = meanA + NPTS;
    float* Tarr  = invA + NPTS;
    float* sq    = Tarr + NPTS;

    // 1) dual GEMM: 2 * (256 m-tiles * 16 n-tiles) tiles, 8 waves/block
    {
        const int tiles = 2 * (NPTS / 16) * (HDIM / 16);
        dgm_dual_gemm<<<tiles / 8, 256, 0, stream>>>(x, W_enc, W_emb, out_x, out_xa);
    }
    // 2) squared norms of x_aux
    dgm_sqnorm<<<NPTS / 8, 256, 0, stream>>>(out_xa, sq);
    // 3) Gram + distance: 64x64 tile per block, async-LDS double-buffered
    {
        const int blocks = (NPTS / 64) * (NPTS / 64);     // 4096
        dgm_gram_dist<<<blocks, 256, 0, stream>>>(out_xa, sq, probs);
    }
    // 4) per-row layernorm stats + entmax threshold
    dgm_rowstats<<<NPTS, 256, 0, stream>>>(probs, gamma, beta, meanA, invA, Tarr);
    // 5) adjacency + logprobs
    dgm_adj<<<NPTS, 256, 0, stream>>>(probs, meanA, invA, Tarr, beta, out_adj, out_lp);
}